// LogSignature_74835510166229
// MI455X (gfx1250) — compile-verified
//
#include <hip/hip_runtime.h>
#include <hip/hip_bf16.h>

#define BATCH 64
#define TLEN  2048
#define NT    2047      // increments
#define DD    6
#define DD2   36
#define DD3   216
#define DD4   1296
#define SIG   1554      // 6+36+216+1296
#define CHUNK 64        // increments per chunk
#define NC    32        // 32*64 = 2048 >= 2047 (last chunk has 63)

typedef float v2f __attribute__((ext_vector_type(2)));
typedef float v8f __attribute__((ext_vector_type(8)));

// ---------------------------------------------------------------------------
// Phase 1: one wave32 per (batch, chunk). Sequential prefix scan builds the
// "exp-corrected" prefix rows p~,q~,r~ in LDS; then the chunk signature's
// levels 2..4 are three matmuls  (rows)^T x X  done with V_WMMA_F32_16X16X4_F32.
// ---------------------------------------------------------------------------
__global__ __launch_bounds__(32)
void sig_chunk_kernel(const float* __restrict__ path, float* __restrict__ csig) {
    extern __shared__ float smem[];
    float* Xs   = smem;                 // [CHUNK][16]  dx rows, cols >=6 zero
    float* Pt   = Xs + CHUNK * 16;      // [CHUNK][8]   p~ rows, cols 6,7 zero
    float* Qt   = Pt + CHUNK * 8;      // [CHUNK][36]  q~ rows
    float* Rt   = Qt + CHUNK * 36;      // [CHUNK][216] r~ rows
    float* pcur = Rt + CHUNK * 216;     // [8]
    float* qcur = pcur + 8;             // [40]
    float* rcur = qcur + 40;            // [224]  (+64 float tail pad below)

    const int lane = threadIdx.x;
    const int b = blockIdx.x / NC;
    const int c = blockIdx.x % NC;
    const int t0 = c * CHUNK;
    const int Lc = min(CHUNK, NT - t0);   // 64, or 63 for the last chunk

    // zero running state (pcur..rcur contiguous: 8+40+224 = 272 floats)
    for (int i = lane; i < 272; i += 32) pcur[i] = 0.0f;

    // load increments dx into LDS (rows >= Lc and cols >= 6 forced to zero,
    // which makes the padded WMMA K/N region contribute exactly 0)
    for (int t = lane; t < CHUNK; t += 32) {
        const float* row = path + ((size_t)b * TLEN + (t0 + t)) * DD;
        #pragma unroll
        for (int j = 0; j < 16; ++j) {
            float v = 0.0f;
            if (t < Lc && j < DD) v = row[DD + j] - row[j];
            Xs[t * 16 + j] = v;
        }
        Pt[t * 8 + 6] = 0.0f; Pt[t * 8 + 7] = 0.0f;
    }
    __syncthreads();

    // sequential scan over the chunk (runs full 64 steps; padded steps have
    // dx == 0 so state is unchanged and matmul contribution is zero)
    for (int t = 0; t < CHUNK; ++t) {
        float dx[6];
        #pragma unroll
        for (int j = 0; j < 6; ++j) dx[j] = Xs[t * 16 + j];

        // level-4 source rows r~ and level-3 state update (uses q,p of t-1)
        for (int idx = lane; idx < DD3; idx += 32) {
            const int k = idx % 6, ij = idx / 6;
            const int i = ij / 6, j = ij - i * 6;
            const float qv = qcur[ij], pv = pcur[i], rv = rcur[idx];
            Rt[t * DD3 + idx] = rv + dx[k] * (0.5f * qv + dx[j] * ((1.0f/6.0f)  * pv + (1.0f/24.0f) * dx[i]));
            rcur[idx]         = rv + dx[k] * (qv        + dx[j] * (0.5f         * pv + (1.0f/6.0f)  * dx[i]));
        }
        __syncthreads();
        // level-3 source rows q~ and level-2 state update (uses p of t-1)
        for (int ij = lane; ij < DD2; ij += 32) {
            const int i = ij / 6, j = ij - i * 6;
            const float pv = pcur[i], qv = qcur[ij];
            Qt[t * DD2 + ij] = qv + dx[j] * (0.5f * pv + (1.0f/6.0f) * dx[i]);
            qcur[ij]         = qv + dx[j] * (pv        + 0.5f        * dx[i]);
        }
        __syncthreads();
        // level-2 source rows p~ and level-1 update
        if (lane < 6) {
            const float pv = pcur[lane];
            Pt[t * 8 + lane] = pv + 0.5f * dx[lane];
            pcur[lane] = pv + dx[lane];
        }
        __syncthreads();
    }

    float* sig = csig + ((size_t)b * NC + c) * SIG;
    if (lane < 6) sig[lane] = pcur[lane];           // level 1 = sum dx

    // ---- WMMA: D[m][n] = sum_t Rows[t][m] * X[t][n], K=CHUNK in steps of 4.
    // A 16x4 f32 layout: lane = 16*(k>=2)+m, vgpr = k&1.  B mirrored (m<->n).
    // C/D f32 16x16: vgpr v holds M=v (lanes 0-15) / M=v+8 (lanes 16-31).
    const int mfr  = lane & 15;
    const int half = lane >> 4;

    // Hoist all B fragments (shared by every tile chain) into registers.
    v2f bfrag[CHUNK / 4];
    #pragma unroll
    for (int kk = 0; kk < CHUNK / 4; ++kk) {
        const int tk = kk * 4 + 2 * half;
        bfrag[kk].x = Xs[tk * 16 + mfr];
        bfrag[kk].y = Xs[(tk + 1) * 16 + mfr];
    }

    // A-fragment loads are deliberately unguarded: rows m >= mvalid read
    // adjacent (finite) LDS inside our allocation and feed only D rows that
    // are discarded at writeout. This keeps EXEC all-ones through the chains.
    for (int tile = 0; tile < 18; ++tile) {
        const float* Amat; int astride, mbase, mvalid, obase;
        if (tile == 0)     { Amat = Pt; astride = 8;   mbase = 0;             mvalid = DD;  obase = 6; }
        else if (tile < 4) { Amat = Qt; astride = DD2; mbase = (tile-1)*16;   mvalid = DD2; obase = 6 + DD2; }
        else               { Amat = Rt; astride = DD3; mbase = (tile-4)*16;   mvalid = DD3; obase = 6 + DD2 + DD3; }
        const int mg = mbase + mfr;

        v8f acc = {0.f, 0.f, 0.f, 0.f, 0.f, 0.f, 0.f, 0.f};
        for (int kk = 0; kk < CHUNK / 4; ++kk) {
            const int tk = kk * 4 + 2 * half;
            v2f a;
            a.x = Amat[(size_t)tk       * astride + mg];
            a.y = Amat[(size_t)(tk + 1) * astride + mg];
            acc = __builtin_amdgcn_wmma_f32_16x16x4_f32(
                false, a, false, bfrag[kk], (short)0, acc, false, false);
        }
        #pragma unroll
        for (int v = 0; v < 8; ++v) {
            const int mm = mbase + v + 8 * half;
            if (mm < mvalid && mfr < 6) sig[obase + mm * 6 + mfr] = acc[v];
        }
    }
}

// ---------------------------------------------------------------------------
// Phase 2: one 256-thread block per batch element. Chen-combine 32 chunk
// signatures, then Horner log-series. Flattened-index tensor products.
// ---------------------------------------------------------------------------
__global__ __launch_bounds__(256)
void combine_kernel(const float* __restrict__ csig, float* __restrict__ out) {
    __shared__ float S[SIG], Bc[SIG], R[SIG];
    const int b = blockIdx.x, tid = threadIdx.x;
    const float* base = csig + (size_t)b * NC * SIG;

    for (int i = tid; i < SIG; i += 256) S[i] = base[i];
    __syncthreads();

    float* S1 = S;  float* S2 = S + 6;  float* S3 = S + 42;  float* S4 = S + 258;
    float* B1 = Bc; float* B2 = Bc + 6; float* B3 = Bc + 42; float* B4 = Bc + 258;
    float* R1 = R;  float* R2 = R + 6;  float* R3 = R + 42;  float* R4 = R + 258;

    for (int c = 1; c < NC; ++c) {
        const float* bs = base + (size_t)c * SIG;
        for (int i = tid; i < SIG; i += 256) Bc[i] = bs[i];
        __syncthreads();
        for (int e = tid; e < DD4; e += 256)
            S4[e] += B4[e] + S1[e/216]*B3[e%216] + S2[e/36]*B2[e%36] + S3[e/6]*B1[e%6];
        __syncthreads();
        for (int e = tid; e < DD3; e += 256)
            S3[e] += B3[e] + S1[e/36]*B2[e%36] + S2[e/6]*B1[e%6];
        __syncthreads();
        for (int e = tid; e < DD2; e += 256)
            S2[e] += B2[e] + S1[e/6]*B1[e%6];
        __syncthreads();
        if (tid < 6) S1[tid] += B1[tid];
        __syncthreads();
    }

    // log(1+S) truncated at depth 4, Horner:  R = c4*S;  R = cn*S + S.R
    for (int i = tid; i < SIG; i += 256) R[i] = -0.25f * S[i];
    __syncthreads();
    const float cs[3] = {1.0f/3.0f, -0.5f, 1.0f};
    for (int it = 0; it < 3; ++it) {
        const float cc = cs[it];
        for (int e = tid; e < DD4; e += 256)
            R4[e] = cc*S4[e] + S1[e/216]*R3[e%216] + S2[e/36]*R2[e%36] + S3[e/6]*R1[e%6];
        __syncthreads();
        for (int e = tid; e < DD3; e += 256)
            R3[e] = cc*S3[e] + S1[e/36]*R2[e%36] + S2[e/6]*R1[e%6];
        __syncthreads();
        for (int e = tid; e < DD2; e += 256)
            R2[e] = cc*S2[e] + S1[e/6]*R1[e%6];
        __syncthreads();
        if (tid < 6) R1[tid] = cc * S1[tid];
        __syncthreads();
    }

    float* o = out + (size_t)b * SIG;
    for (int i = tid; i < SIG; i += 256) o[i] = R[i];
}

extern "C" void kernel_launch(void* const* d_in, const int* in_sizes, int n_in,
                              void* d_out, int out_size, void* d_ws, size_t ws_size,
                              hipStream_t stream) {
    (void)in_sizes; (void)n_in; (void)out_size; (void)ws_size;
    const float* path = (const float*)d_in[0];
    float* out  = (float*)d_out;
    float* csig = (float*)d_ws;   // needs 64*32*1554*4 = ~12.7 MB scratch

    // +64 floats of tail pad so deliberately-unguarded A reads stay in-bounds
    const size_t shbytes =
        (size_t)(CHUNK*16 + CHUNK*8 + CHUNK*36 + CHUNK*216 + 272 + 64) * sizeof(float);

    sig_chunk_kernel<<<dim3(BATCH * NC), dim3(32), shbytes, stream>>>(path, csig);
    combine_kernel<<<dim3(BATCH), dim3(256), 0, stream>>>(csig, out);
}